// blocks_InfoNCE_PCA_47665547051141
// MI455X (gfx1250) — compile-verified
//
#include <hip/hip_runtime.h>
#include <hip/hip_bf16.h>

// ---------------------------------------------------------------------------
// Problem constants (from reference): B=96, C=768, H=W=20 -> N=400 tokens.
// ---------------------------------------------------------------------------
#define BATCH   96
#define CHAN    768
#define NPIX    400           // 20*20
#define NTILE   25            // 400 / 16
#define NGRP    5             // col-groups of 5 tiles: 25 = 5*5 -> wave does 16x80
#define JOBS    (NTILE*NGRP)  // 125 wave-jobs per batch
#define EMBD    (3*NPIX)      // 1200
#define NITER   128           // subspace-iteration sweeps

typedef __attribute__((ext_vector_type(2))) float v2f;
typedef __attribute__((ext_vector_type(8))) float v8f;

// ---------------------------------------------------------------------------
// Kernel 1: per-(batch,channel) mean over the 400 spatial positions.
// One wave32 per contiguous row -> coalesced, shuffle-reduced.
// ---------------------------------------------------------------------------
__global__ __launch_bounds__(256) void mean_kernel(const float* __restrict__ feat,
                                                   float* __restrict__ mu) {
    const int warp = (blockIdx.x * 256 + threadIdx.x) >> 5;
    const int lane = threadIdx.x & 31;
    if (warp >= BATCH * CHAN) return;
    const float* p = feat + (size_t)warp * NPIX;
    float s = 0.f;
    for (int n = lane; n < NPIX; n += 32) s += p[n];
    #pragma unroll
    for (int off = 16; off >= 1; off >>= 1) s += __shfl_down(s, off, 32);
    if (lane == 0) mu[warp] = s * (1.0f / NPIX);
}

// ---------------------------------------------------------------------------
// Kernel 2: Gram matrix G_b = Xc * Xc^T  (400x400, K=768) via fp32 WMMA.
//   X stored as feat[b][c][n] (n contiguous); Xc[m][k] = feat[b][k][m]-mu[k].
//   Each wave computes a 16x80 strip: one A fragment feeds 5 B fragments /
//   5 accumulators per k-step (V_WMMA_F32_16X16X4_F32), ~6.7 FLOP/B from L2.
//
//   f32 16x16x4 fragment layout (ISA 7.12.2):
//     A (16x4): lanes 0-15 = rows M, VGPR0/1 = {K=0,K=1}; lanes 16-31 = {K=2,K=3}
//     B (4x16): lanes 0-15 = cols N, same K split across lane halves
//     C/D:      VGPR r, lanes 0-15 -> (M=r, N=lane); lanes 16-31 -> (M=r+8).
// ---------------------------------------------------------------------------
__global__ __launch_bounds__(256) void gram_kernel(const float* __restrict__ feat,
                                                   const float* __restrict__ mu,
                                                   float* __restrict__ gram) {
    const int b   = blockIdx.y;
    const int job = blockIdx.x * 8 + (threadIdx.x >> 5);    // 8 waves / block
    if (job >= JOBS) return;                                 // wave-uniform exit
    const int tm  = job / NGRP;          // row tile 0..24
    const int cg  = job % NGRP;          // column group 0..4 (80 cols each)
    const int lane = threadIdx.x & 31;
    const int half = lane >> 4;          // K-group selector
    const int l16  = lane & 15;

    const float* Y  = feat + (size_t)b * CHAN * NPIX;  // Y[k][n], n contiguous
    const float* MU = mu   + b * CHAN;
    const int m     = tm * 16 + l16;
    const int nbase = cg * 80 + l16;

    v8f acc[NGRP];
    #pragma unroll
    for (int g = 0; g < NGRP; ++g) acc[g] = (v8f){0.f,0.f,0.f,0.f,0.f,0.f,0.f,0.f};

    #pragma unroll 2
    for (int k = 0; k < CHAN; k += 4) {
        const int k0 = k + 2 * half;
        const float mu0 = MU[k0];
        const float mu1 = MU[k0 + 1];
        const float* r0 = Y + (size_t)k0 * NPIX;
        const float* r1 = r0 + NPIX;
        v2f a;
        a.x = r0[m] - mu0;
        a.y = r1[m] - mu1;
        #pragma unroll
        for (int g = 0; g < NGRP; ++g) {
            v2f bb;
            bb.x = r0[nbase + g * 16] - mu0;
            bb.y = r1[nbase + g * 16] - mu1;
            acc[g] = __builtin_amdgcn_wmma_f32_16x16x4_f32(false, a, false, bb,
                                                           (short)0, acc[g],
                                                           false, false);
        }
    }

    float* Gout = gram + (size_t)b * NPIX * NPIX;
    #pragma unroll
    for (int g = 0; g < NGRP; ++g) {
        #pragma unroll
        for (int r = 0; r < 8; ++r) {
            const int row = tm * 16 + r + 8 * half;
            Gout[(size_t)row * NPIX + cg * 80 + g * 16 + l16] = acc[g][r];
        }
    }
}

// ---------------------------------------------------------------------------
// Kernel 3: top-3 eigenpairs of each 400x400 Gram via orthogonal iteration,
// one workgroup (256 threads) per batch.  G is symmetric, so the matvec is
// done column-major: lanes read G[k*400 + i] contiguously (coalesced) and
// v[j][k] is an LDS broadcast.
// ---------------------------------------------------------------------------
__device__ __forceinline__ float block_dot(const float* a, const float* c,
                                           float* red) {
    float s = 0.f;
    for (int i = threadIdx.x; i < NPIX; i += 256) s += a[i] * c[i];
    red[threadIdx.x] = s;
    __syncthreads();
    #pragma unroll
    for (int off = 128; off >= 1; off >>= 1) {
        if ((int)threadIdx.x < off) red[threadIdx.x] += red[threadIdx.x + off];
        __syncthreads();
    }
    const float r = red[0];
    __syncthreads();
    return r;
}

__global__ __launch_bounds__(256) void eig_kernel(const float* __restrict__ gram,
                                                  float* __restrict__ emb) {
    __shared__ float v[3][NPIX];
    __shared__ float y[3][NPIX];
    __shared__ float red[256];
    __shared__ float absr[256];
    __shared__ int   idxr[256];

    const int b   = blockIdx.x;
    const int tid = threadIdx.x;
    const float* G = gram + (size_t)b * NPIX * NPIX;
    const int i1 = tid;                 // column owned by this thread
    const int i2 = tid + 256;           // second column (valid if < 400)
    const bool has2 = (i2 < NPIX);

    // deterministic pseudo-random init
    for (int i = tid; i < NPIX; i += 256) {
        #pragma unroll
        for (int j = 0; j < 3; ++j) {
            unsigned h = (unsigned)i * 2654435761u + (unsigned)(j + 1) * 40503u + 12345u;
            h ^= h >> 13; h *= 0x5bd1e995u; h ^= h >> 15;
            v[j][i] = (float)(h & 0xFFFFFF) * (2.0f / 16777216.0f) - 1.0f;
        }
    }
    __syncthreads();

    for (int it = 0; it <= NITER; ++it) {
        // y_j = G v_j for j=0..2, column-major (coalesced) using symmetry
        {
            float a0 = 0.f, a1 = 0.f, a2 = 0.f;
            float c0 = 0.f, c1 = 0.f, c2 = 0.f;
            for (int k = 0; k < NPIX; ++k) {
                const float* rowk = G + (size_t)k * NPIX;
                const float vk0 = v[0][k], vk1 = v[1][k], vk2 = v[2][k];
                const float g1 = rowk[i1];
                a0 += g1 * vk0; a1 += g1 * vk1; a2 += g1 * vk2;
                if (has2) {
                    const float g2 = rowk[i2];
                    c0 += g2 * vk0; c1 += g2 * vk1; c2 += g2 * vk2;
                }
            }
            y[0][i1] = a0; y[1][i1] = a1; y[2][i1] = a2;
            if (has2) { y[0][i2] = c0; y[1][i2] = c1; y[2][i2] = c2; }
        }
        __syncthreads();
        if (it == NITER) break;   // final y kept raw for Rayleigh quotients

        // Gram-Schmidt (keeps descending eigenvalue order)
        for (int j = 0; j < 3; ++j) {
            for (int l = 0; l < j; ++l) {
                const float d = block_dot(y[j], y[l], red);
                for (int i = tid; i < NPIX; i += 256) y[j][i] -= d * y[l][i];
                __syncthreads();
            }
            const float nrm = sqrtf(block_dot(y[j], y[j], red)) + 1e-30f;
            const float inv = 1.0f / nrm;
            for (int i = tid; i < NPIX; i += 256) v[j][i] = y[j][i] * inv;
            __syncthreads();
        }
    }

    // Rayleigh quotients: lambda_j = v_j . (G v_j)   (v orthonormal)
    float lam[3], sv[3];
    for (int j = 0; j < 3; ++j) {
        lam[j] = block_dot(v[j], y[j], red);
        sv[j]  = sqrtf(fmaxf(lam[j], 0.f));
    }
    // ||emb||^2 = lam0+lam1+lam2  ->  normalization is free
    const float invn = rsqrtf(fmaxf(lam[0] + lam[1] + lam[2], 1e-30f));

    // sklearn svd_flip: sign of max-|.| entry of each eigenvector column
    float* out = emb + (size_t)b * EMBD;
    for (int j = 0; j < 3; ++j) {
        float ba = -1.f; int bi = 0;
        for (int i = tid; i < NPIX; i += 256) {
            const float a = fabsf(v[j][i]);
            if (a > ba) { ba = a; bi = i; }
        }
        absr[tid] = ba; idxr[tid] = bi;
        __syncthreads();
        #pragma unroll
        for (int off = 128; off >= 1; off >>= 1) {
            if (tid < off) {
                const bool take = (absr[tid + off] > absr[tid]) ||
                                  (absr[tid + off] == absr[tid] && idxr[tid + off] < idxr[tid]);
                if (take) { absr[tid] = absr[tid + off]; idxr[tid] = idxr[tid + off]; }
            }
            __syncthreads();
        }
        const float sgn = (v[j][idxr[0]] >= 0.f) ? 1.0f : -1.0f;
        __syncthreads();
        const float scl = sgn * sv[j] * invn;
        for (int i = tid; i < NPIX; i += 256) out[j * NPIX + i] = v[j][i] * scl;
    }
}

// ---------------------------------------------------------------------------
// Kernel 4: logits = scale * E @ E^T (96x96, K=1200; emb1==emb2 per the
// reference's own bug).  6x6 = 36 tiles of 16x16, one wave each, fp32 WMMA.
// ---------------------------------------------------------------------------
__global__ __launch_bounds__(128) void logits_kernel(const float* __restrict__ emb,
                                                     const float* __restrict__ scale_p,
                                                     float* __restrict__ logits) {
    const int tile = blockIdx.x * 4 + (threadIdx.x >> 5);   // 4 waves / block
    const int tm   = tile / 6;
    const int tn   = tile % 6;
    const int lane = threadIdx.x & 31;
    const int half = lane >> 4;
    const int l16  = lane & 15;
    const float scale = scale_p[0];

    const float* Am = emb + (size_t)(tm * 16 + l16) * EMBD;  // row m of E
    const float* Bn = emb + (size_t)(tn * 16 + l16) * EMBD;  // row n of E

    v8f acc = {0.f,0.f,0.f,0.f,0.f,0.f,0.f,0.f};
    #pragma unroll 4
    for (int k = 0; k < EMBD; k += 4) {
        const int k0 = k + 2 * half;
        v2f a, bb;
        a.x  = Am[k0];     a.y  = Am[k0 + 1];
        bb.x = Bn[k0];     bb.y = Bn[k0 + 1];
        acc = __builtin_amdgcn_wmma_f32_16x16x4_f32(false, a, false, bb,
                                                    (short)0, acc, false, false);
    }
    #pragma unroll
    for (int r = 0; r < 8; ++r) {
        const int row = tm * 16 + r + 8 * half;
        logits[row * BATCH + tn * 16 + l16] = scale * acc[r];
    }
}

// ---------------------------------------------------------------------------
// Kernel 5: loss. Logits matrix is exactly symmetric (same buffer, same
// summation order), so 0.5*(CE(L)+CE(L^T)) == mean_i(lse(L[i,:]) - L[i,i]).
// ---------------------------------------------------------------------------
__global__ __launch_bounds__(128) void loss_kernel(const float* __restrict__ logits,
                                                   float* __restrict__ out) {
    __shared__ float red[128];
    const int tid = threadIdx.x;
    float part = 0.f;
    if (tid < BATCH) {
        const float* row = logits + tid * BATCH;
        float mx = row[0];
        for (int k = 1; k < BATCH; ++k) mx = fmaxf(mx, row[k]);
        float se = 0.f;
        for (int k = 0; k < BATCH; ++k) se += expf(row[k] - mx);
        part = (mx + logf(se)) - row[tid];
    }
    red[tid] = part;
    __syncthreads();
    #pragma unroll
    for (int off = 64; off >= 1; off >>= 1) {
        if (tid < off) red[tid] += red[tid + off];
        __syncthreads();
    }
    if (tid == 0) out[0] = red[0] * (1.0f / BATCH);
}

// ---------------------------------------------------------------------------
// Launcher.  Workspace layout (floats):
//   mu     [96*768]        =    73,728
//   gram   [96*400*400]    = 15,360,000
//   emb    [96*1200]       =   115,200
//   logits [96*96]         =     9,216
//   total ~62.2 MB
// ---------------------------------------------------------------------------
extern "C" void kernel_launch(void* const* d_in, const int* in_sizes, int n_in,
                              void* d_out, int out_size, void* d_ws, size_t ws_size,
                              hipStream_t stream) {
    const float* feat1 = (const float*)d_in[0];   // [96,768,20,20]
    const float* scale = (const float*)d_in[2];   // scalar logit_scale
    float* out = (float*)d_out;

    float* ws     = (float*)d_ws;
    float* mu     = ws;
    float* gram   = mu   + BATCH * CHAN;
    float* emb    = gram + (size_t)BATCH * NPIX * NPIX;
    float* logits = emb  + BATCH * EMBD;

    // 1) per-channel means: one wave per (b,c) row
    {
        const int rows = BATCH * CHAN;
        const int blocks = (rows * 32 + 255) / 256;
        mean_kernel<<<blocks, 256, 0, stream>>>(feat1, mu);
    }
    // 2) Gram matrices via fp32 WMMA: 16x80 strip per wave, grid.y = batch
    {
        dim3 grid((JOBS + 7) / 8, BATCH);
        gram_kernel<<<grid, 256, 0, stream>>>(feat1, mu, gram);
    }
    // 3) top-3 eigenpairs + normalized embedding: one block per batch
    eig_kernel<<<BATCH, 256, 0, stream>>>(gram, emb);
    // 4) 96x96 similarity logits via fp32 WMMA (36 tiles, 4 waves/block)
    logits_kernel<<<9, 128, 0, stream>>>(emb, scale, logits);
    // 5) symmetric InfoNCE loss -> scalar
    loss_kernel<<<1, 128, 0, stream>>>(logits, out);
}